// CRF_33870112096703
// MI455X (gfx1250) — compile-verified
//
#include <hip/hip_runtime.h>
#include <hip/hip_bf16.h>

#define NUM_TAG   24
#define TP        26          // NUM_TAG + 2
#define START_ID  24
#define STOP_ID   25
#define PAD_ID    23
#define MAX_L     512
#define WPB       4           // waves (= sequences) per block
#define BLOCK_T   (WPB * 32)

__device__ __forceinline__ float bcast_lane(float v, int srcLane) {
  return __builtin_bit_cast(float,
      __builtin_amdgcn_readlane(__builtin_bit_cast(int, v), srcLane));
}

__global__ __launch_bounds__(BLOCK_T)
void viterbi_crf_kernel(const float* __restrict__ feats,   // [B, L, 24]
                        const int*   __restrict__ mask,    // [B, L] prefix mask
                        const float* __restrict__ trans,   // [26, 26], trans[next, prev]
                        float* __restrict__ out_paths,     // [B, L] (tags as float)
                        float* __restrict__ out_scores,    // [B]
                        int B, int L)
{
  __shared__ float         s_trans[TP * TP];                       // 2704 B
  __shared__ unsigned char s_bp[WPB][MAX_L * NUM_TAG + 32];        // 12 KB + slack per wave
  __shared__ unsigned char s_path[WPB][MAX_L];

  const int tid  = threadIdx.x;
  const int lane = tid & 31;
  const int wv   = tid >> 5;
  int b = blockIdx.x * WPB + wv;
  if (b >= B) b = 0;   // duplicate of seq 0: identical deterministic writes, benign

  // ---- Stage transition table into LDS via gfx1250 async global->LDS DMA ----
  for (int i = tid; i < TP * TP; i += BLOCK_T) {
    unsigned int       ldsOff = (unsigned int)(unsigned long long)(const void*)&s_trans[i];
    unsigned long long gAddr  = (unsigned long long)(const void*)(trans + i);
    asm volatile("global_load_async_to_lds_b32 %0, %1, off"
                 :: "v"(ldsOff), "v"(gAddr)
                 : "memory");
  }

  const int*   mb = mask  + (size_t)b * L;
  const float* fb = feats + (size_t)b * L * NUM_TAG;

  // ---- Sequence length = popcount of prefix mask (coalesced, no serial loads) ----
  int psum = 0;
#pragma unroll
  for (int k = 0; k < MAX_L / 32; ++k) {
    const int tt = k * 32 + lane;
    if (tt < L) psum += mb[tt];
  }
#pragma unroll
  for (int off = 16; off > 0; off >>= 1) psum += __shfl_xor(psum, off, 32);
  const int len = __builtin_amdgcn_readfirstlane(psum);   // 1 <= len <= L

  asm volatile("s_wait_asynccnt 0" ::: "memory");
  __syncthreads();

  // lane = next-tag (lanes 24..31 are padding lanes)
  const int row = (lane < NUM_TAG) ? lane : 0;

  float trow[NUM_TAG];                       // trans[row, p] kept in VGPRs
#pragma unroll
  for (int p = 0; p < NUM_TAG; ++p) trow[p] = s_trans[row * TP + p];
  const float tStart = s_trans[row * TP + START_ID];
  const float tStop  = s_trans[STOP_ID * TP + row];

  unsigned char* bps  = &s_bp[wv][0];
  unsigned char* pbuf = &s_path[wv][0];

  float fv = tStart + fb[row];               // t = 0 (mask[0] == 1 guaranteed)

  // Software-pipelined feat stream: feat_buf holds feats[t] while the load for
  // t+1 is in flight; the wait lands at the end of the ~120-VALU step body.
  float feat_buf = fb[NUM_TAG + row];        // feats[t=1] (L >= 2 always here)

  for (int t = 1; t < len; ++t) {
    const int   tn   = (t + 1 < L) ? (t + 1) : t;              // in-bounds clamp
    const float next = fb[(size_t)tn * NUM_TAG + row];         // issue early
    __builtin_prefetch(fb + (size_t)(t + 8) * NUM_TAG + row, 0, 3); // global_prefetch_b8

    float v[NUM_TAG];
    int   id[NUM_TAG];
#pragma unroll
    for (int p = 0; p < NUM_TAG; ++p) {      // 24 independent readlane+add
      v[p]  = bcast_lane(fv, p) + trow[p];
      id[p] = p;
    }
    // argmax tree: left operand always has lower indices, so `>=` keeps the
    // FIRST maximum (matches jnp.argmax tie-break) with log-depth chains.
#pragma unroll
    for (int s = 1; s < NUM_TAG; s <<= 1) {
#pragma unroll
      for (int i = 0; i + s < NUM_TAG; i += 2 * s) {
        const bool keep = v[i] >= v[i + s];
        v[i]  = keep ? v[i]  : v[i + s];
        id[i] = keep ? id[i] : id[i + s];
      }
    }
    fv = v[0] + feat_buf;
    // lanes 24..31 write into row t+1 cols 0..7: those bytes are rewritten by
    // the next step before ever being read (LDS ops of a wave are in order);
    // the +32 slack covers the final step's spill.
    bps[t * NUM_TAG + lane] = (unsigned char)id[0];
    feat_buf = next;                         // wait for the t+1 load lands here
  }

  // terminal score + full-wave argmax (first max wins on ties)
  float term = (lane < NUM_TAG) ? (fv + tStop) : -3.0e38f;
  int   bidx = lane;
#pragma unroll
  for (int off = 16; off > 0; off >>= 1) {
    const float ov = __shfl_xor(term, off, 32);
    const int   oi = __shfl_xor(bidx, off, 32);
    if (ov > term || (ov == term && oi < bidx)) { term = ov; bidx = oi; }
  }

  __syncthreads();                           // order bp writes before backtrace reads

  if (lane == 0) {                           // serial backtrace over LDS (low latency)
    int cur = bidx;
    for (int tt = len - 1; tt >= 1; --tt) {
      pbuf[tt] = (unsigned char)cur;
      cur = bps[tt * NUM_TAG + cur];
    }
    pbuf[0] = (unsigned char)cur;
  }

  __syncthreads();                           // order path writes before writeout

  const size_t bL = (size_t)b * L;
#pragma unroll
  for (int k = 0; k < MAX_L / 32; ++k) {     // coalesced float writeout
    const int tt = k * 32 + lane;
    if (tt < L) {
      const float val = (tt < len) ? (float)pbuf[tt] : (float)PAD_ID;
      out_paths[bL + tt] = val;
    }
  }
  if (lane == 0) out_scores[b] = term;
}

extern "C" void kernel_launch(void* const* d_in, const int* in_sizes, int n_in,
                              void* d_out, int out_size, void* d_ws, size_t ws_size,
                              hipStream_t stream) {
  (void)n_in; (void)d_ws; (void)ws_size;
  const float* feats = (const float*)d_in[0];
  const int*   mask  = (const int*)d_in[1];
  const float* trans = (const float*)d_in[2];
  float*       out   = (float*)d_out;

  // out_size = B*(L+1), mask elements = B*L  =>  B = out_size - maskN
  const int maskN = in_sizes[1];
  const int B = out_size - maskN;
  const int L = maskN / B;

  float* out_paths  = out;
  float* out_scores = out + (size_t)B * L;

  const int blocks = (B + WPB - 1) / WPB;
  viterbi_crf_kernel<<<blocks, BLOCK_T, 0, stream>>>(feats, mask, trans,
                                                     out_paths, out_scores, B, L);
}